// ViterbiLoss_74706661147198
// MI455X (gfx1250) — compile-verified
//
#include <hip/hip_runtime.h>

// Problem constants from the reference
#define B_         128
#define T_         256
#define K_         48
#define KH_        24            // half of K (2-way split of the i-reduction)
#define TSZ        2304          // K*K elements per (b,t) tile
#define TILE_BYTES 9216          // TSZ * 4
#define START_TAG_ 46
#define STOP_TAG_  47
#define DEPTH      24            // TDM tiles in flight (24*9216 = 221 KB LDS)

typedef unsigned int v4u __attribute__((ext_vector_type(4)));
typedef unsigned int v8u __attribute__((ext_vector_type(8)));

struct FwdLds {
    float tiles[DEPTH][TSZ];
    float scores[K_];
    float pm[2][K_];             // partial max  per (half, column)
    float ps[2][K_];             // partial expsum per (half, column)
};

// Issue one TDM copy of a full 9216B tile (1-D, 2304 x 4B elements) into LDS.
// D# group0: count=1, lds_addr, global_addr[56:0], type=2.  Non-temporal: the
// feature stream is read exactly once, keep it out of L2's 192MB.
__device__ __forceinline__ void tdm_issue(unsigned lds_off,
                                          unsigned long long gaddr,
                                          v8u g1) {
    v4u g0;
    g0.x = 1u;                                                // count=1, user mode
    g0.y = lds_off;                                           // LDS byte address
    g0.z = (unsigned)gaddr;                                   // global_addr[31:0]
    g0.w = ((unsigned)((gaddr >> 32) & 0x01FFFFFFull))        // global_addr[56:32]
           | 0x80000000u;                                     // type=2 at bits[127:126]
    asm volatile("tensor_load_to_lds %0, %1 th:TH_LOAD_NT" :: "s"(g0), "s"(g1) : "memory");
}

__global__ __launch_bounds__(96)
void crf_forward_kernel(const float* __restrict__ features,
                        const long long* __restrict__ lengths,
                        float* __restrict__ finals) {
    __shared__ FwdLds lds;

    const int b   = blockIdx.x;
    const int tid = threadIdx.x;
    const int h   = tid / K_;            // which half of the i-range (0 or 1)
    const int j   = tid - h * K_;        // column owned by this thread
    const int len = (int)lengths[b];
    const float* fb = features + (size_t)b * T_ * TSZ;

    // init: scores = features[b, 0, START_TAG, :]
    if (tid < K_) lds.scores[tid] = fb[START_TAG_ * K_ + tid];

    // Scalar (wave-uniform) predicate: only wave 0 issues TDM / waits TENSORcnt.
    // Must be a *branch*, not EXEC masking, since TDM ignores EXEC.
    const bool issuer = (__builtin_amdgcn_readfirstlane((int)threadIdx.x) < 32);

    const unsigned long long gbase = (unsigned long long)(uintptr_t)fb;

    // D# group1 (constant): data_size=4B; 1-D tile: tile_dim0=2304, tile_dim1=0;
    // tensor_dim0=2304, tensor_dim1=1, tensor_dim0_stride=2304.
    const v8u g1 = { 0x00020000u,   // wg_mask=0 | data_size=2(4B)
                     0x09000000u,   // tensor_dim0[15:0]=2304 in [31:16]
                     0x00010000u,   // tensor_dim0[31:16]=0 | tensor_dim1[15:0]=1
                     0x09000000u,   // tensor_dim1[31:16]=0 | tile_dim0=2304
                     0x00000000u,   // tile_dim1=0, tile_dim2=0
                     0x00000900u,   // tensor_dim0_stride[31:0]=2304
                     0x00000000u,   // stride0[47:32]=0 | stride1[15:0]=0
                     0x00000000u }; // stride1[47:16]=0

    // Prologue: fill the pipeline with DEPTH tiles (t = 1..DEPTH).
    if (issuer) {
        #pragma unroll
        for (int p = 0; p < DEPTH; ++p) {
            int tp = p + 1;
            tdm_issue((unsigned)(p * TILE_BYTES),
                      gbase + (unsigned long long)tp * TILE_BYTES, g1);
        }
    }

    const int nsteps = len - 1;              // recursion runs t = 1 .. len-1

    for (int s = 0; s < nsteps; ++s) {
        // Oldest of the DEPTH outstanding TDM loads (slot s%DEPTH) is complete
        // once TENSORcnt <= DEPTH-1 (same-wave tensor ops complete in order).
        if (issuer) __builtin_amdgcn_s_wait_tensorcnt(DEPTH - 1);
        __syncthreads();                     // barrier A: tile + scores visible

        const int slot = s % DEPTH;
        const float* tile = &lds.tiles[slot][0];

        // Pass 1: each thread reduces its 24-row half of column j.
        {
            float f[KH_];
            float m = -3.402823466e+38f;
            const int i0 = h * KH_;
            #pragma unroll
            for (int i = 0; i < KH_; ++i) {
                float v = tile[(i0 + i) * K_ + j] + lds.scores[i0 + i];
                f[i] = v;
                m = fmaxf(m, v);
            }
            float ssum = 0.0f;
            #pragma unroll
            for (int i = 0; i < KH_; ++i) ssum += __expf(f[i] - m);
            lds.pm[h][j] = m;
            lds.ps[h][j] = ssum;
        }
        __syncthreads();                     // barrier B: partials published,
                                             // everyone done with tile & scores
        // Pass 2: merge the two (max,sum) partials; owner writes new score.
        if (tid < K_) {
            float m0 = lds.pm[0][tid], m1 = lds.pm[1][tid];
            float s0 = lds.ps[0][tid], s1 = lds.ps[1][tid];
            float m  = fmaxf(m0, m1);
            float ss = s0 * __expf(m0 - m) + s1 * __expf(m1 - m);
            lds.scores[tid] = m + __logf(ss);   // read next iter after barrier A
        }

        // Refill the slot we just consumed (clamped dummy loads past the end
        // keep the outstanding count pinned at DEPTH so the wait imm is const).
        if (issuer) {
            int tp = s + 1 + DEPTH;
            if (tp > T_ - 1) tp = T_ - 1;
            tdm_issue((unsigned)(slot * TILE_BYTES),
                      gbase + (unsigned long long)tp * TILE_BYTES, g1);
        }
    }

    if (issuer) __builtin_amdgcn_s_wait_tensorcnt(0);   // drain before exit
    __syncthreads();
    if (tid == 0) finals[b] = lds.scores[STOP_TAG_];
}

__global__ __launch_bounds__(256)
void gold_kernel(const float* __restrict__ features,
                 const long long* __restrict__ targets,
                 const long long* __restrict__ lengths,
                 float* __restrict__ golds) {
    __shared__ float red[256];
    const int b = blockIdx.x;
    const int t = threadIdx.x;
    const int len = (int)lengths[b];

    float g = 0.0f;
    if (t < len) {
        const long long tgt = targets[(size_t)b * T_ + t];
        g = features[(size_t)(b * T_ + t) * TSZ + (size_t)tgt];
    }
    red[t] = g;
    __syncthreads();
    #pragma unroll
    for (int off = 128; off > 0; off >>= 1) {
        if (t < off) red[t] += red[t + off];
        __syncthreads();
    }
    if (t == 0) golds[b] = red[0];
}

__global__ __launch_bounds__(128)
void finalize_kernel(const float* __restrict__ finals,
                     const float* __restrict__ golds,
                     float* __restrict__ out) {
    __shared__ float red[128];
    const int t = threadIdx.x;
    red[t] = finals[t] - golds[t];
    __syncthreads();
    #pragma unroll
    for (int off = 64; off > 0; off >>= 1) {
        if (t < off) red[t] += red[t + off];
        __syncthreads();
    }
    if (t == 0) out[0] = red[0];
}

extern "C" void kernel_launch(void* const* d_in, const int* in_sizes, int n_in,
                              void* d_out, int out_size, void* d_ws, size_t ws_size,
                              hipStream_t stream) {
    const float*     features = (const float*)d_in[0];
    const long long* targets  = (const long long*)d_in[1];
    const long long* lengths  = (const long long*)d_in[2];

    float* finals = (float*)d_ws;        // [0, 128)  per-b terminal scores
    float* golds  = finals + B_;         // [128,256) per-b gold partial sums

    gold_kernel<<<B_, 256, 0, stream>>>(features, targets, lengths, golds);
    crf_forward_kernel<<<B_, 96, 0, stream>>>(features, lengths, finals);
    finalize_kernel<<<1, 128, 0, stream>>>(finals, golds, (float*)d_out);
}